// ProjectLayerTS_75900662055115
// MI455X (gfx1250) — compile-verified
//
#include <hip/hip_runtime.h>
#include <stdint.h>

typedef unsigned int u32;
typedef unsigned long long u64;
typedef __attribute__((ext_vector_type(4))) u32 sdesc4;
typedef __attribute__((ext_vector_type(8))) u32 sdesc8;

#define NVIEW 5
#define NJ 15
#define HH 128
#define WW 240
#define PLANE (HH * WW)          // 30720 floats = 122,880 B per (view,joint) plane
#define FVX 253
#define FVY 253
#define FVZ 64
#define NP 10
#define XSPLIT 16                // split the 64 x-slices into 16 chunks of 4
#define XCHUNK 4
#define CUBES_ELEMS (NP * NJ * 64 * 64 * 64)   // 39,321,600

__device__ __forceinline__ int imin(int a, int b) { return a < b ? a : b; }
__device__ __forceinline__ int imax(int a, int b) { return a > b ? a : b; }

// Issue a TDM copy of one 122,880-byte heatmap plane (30720 dwords, contiguous)
// from global memory into LDS. D# built per CDNA5 ISA:
//   group0: count=1 | lds_addr | global_addr[56:0] | type=2
//   group1: data_size=4B, tensor_dim0=tile_dim0=30720, tensor_dim1=tile_dim1=1,
//           tensor_dim0_stride=30720, no multicast/pad/iterate.
// Tracked by TENSORcnt; 2D tensor -> 2 SGPR groups (VADDR2/3 omitted = NULL).
__device__ __forceinline__ void tdm_load_plane(const float* gsrc, const float* lds_dst) {
  u64 ga = (u64)(uintptr_t)gsrc;
  u32 lds_off = (u32)(uintptr_t)(const void*)lds_dst;  // low 32 bits of AS(3) cast = LDS byte offset
  sdesc4 d0;
  d0[0] = 1u;                                                  // count = 1 (valid)
  d0[1] = lds_off;                                             // lds_addr
  d0[2] = (u32)ga;                                             // global_addr[31:0]
  d0[3] = (u32)((ga >> 32) & 0x1FFFFFFull) | (2u << 30);       // global_addr[56:32] | type=2
  sdesc8 d1;
  d1[0] = 2u << 16;                                            // data_size = 2 (4 bytes); mask/pad/iterate = 0
  d1[1] = ((u32)PLANE & 0xFFFFu) << 16;                        // tensor_dim0[15:0] at bits[63:48]
  d1[2] = ((u32)PLANE >> 16) | (1u << 16);                     // tensor_dim0[31:16] | tensor_dim1[15:0]=1
  d1[3] = ((u32)PLANE) << 16;                                  // tensor_dim1[31:16]=0 | tile_dim0=30720
  d1[4] = 1u;                                                  // tile_dim1=1, tile_dim2=0
  d1[5] = (u32)PLANE;                                          // tensor_dim0_stride[31:0]
  d1[6] = 0u;                                                  // stride0 hi16 / stride1 lo16
  d1[7] = 0u;                                                  // stride1 hi32
  asm volatile("tensor_load_to_lds %0, %1" :: "s"(d0), "s"(d1) : "memory");
}

__global__ __launch_bounds__(1024, 1) void ProjectLayerTS_kernel(
    const float* __restrict__ heatmaps,   // (1, 5, 15, 128, 240)
    const float* __restrict__ grids,      // (5, 253, 253, 64, 2)
    const float* __restrict__ pcs,        // (10, 7)
    const int*   __restrict__ indexp,     // scalar
    float*       __restrict__ out)        // 39,321,600 cubes + 30 offsets
{
  extern __shared__ float lds[];          // 2 * PLANE floats (double buffer)

  const int bx    = blockIdx.x;           // NP*NJ*XSPLIT = 2400
  const int chunk = bx % XSPLIT;
  const int pj    = bx / XSPLIT;
  const int j     = pj % NJ;
  const int p     = pj / NJ;
  const int tid   = threadIdx.x;

  // ---- per-proposal scalars (uniform; every thread recomputes, trivially cheap) ----
  const float px = pcs[p * 7 + 0], py = pcs[p * 7 + 1], pz = pcs[p * 7 + 2];
  const float s5 = pcs[p * 7 + 5], s6 = pcs[p * 7 + 6];
  // ctl = round(pos*SCALE + BIAS), SCALE=252/8000 (xy), 63/2000 (z); BIAS=94.5,94.5,-31.5
  const int cx = (int)rintf(px * (252.0f / 8000.0f) + 94.5f);   // rintf = RNE, matches jnp.round
  const int cy = (int)rintf(py * (252.0f / 8000.0f) + 94.5f);
  const int cz = (int)rintf(pz * (63.0f / 2000.0f) - 31.5f);
  int mx = (int)((1.0f - s5) * 0.5f * 63.0f); mx = imax(mx, 0);
  int my = (int)((1.0f - s6) * 0.5f * 63.0f); my = imax(my, 0);
  const int sx = imax(cx + mx, 0), sy = imax(cy + my, 0), sz = imax(cz, 0);
  const int ex = imin(cx + 64 - mx, FVX), ey = imin(cy + 64 - my, FVY), ez = imin(cz + 64, FVZ);
  const bool person_ok = (sx < ex) && (sy < ey) && (sz < ez);

  // ---- offset output (written by one block per proposal) ----
  if (j == 0 && chunk == 0 && tid == 0) {
    out[CUBES_ELEMS + p * 3 + 0] = (float)cx * (8000.0f / 252.0f) - 3000.0f;
    out[CUBES_ELEMS + p * 3 + 1] = (float)cy * (8000.0f / 252.0f) - 3000.0f;
    out[CUBES_ELEMS + p * 3 + 2] = (float)cz * (2000.0f / 63.0f);
  }

  const int idx = indexp[0];
  const float* hmb = heatmaps + (u64)idx * NVIEW * NJ * PLANE + (u64)j * PLANE;

  // thread -> voxel mapping: z = tid&63 (fixed per thread; coalesced along z),
  // 16 voxels per thread spanning (x within chunk, y).
  const int z   = tid & 63;
  const int fzi = cz + z;
  const bool vz = (fzi >= sz) && (fzi < ez);
  const int fzc = imin(imax(fzi, 0), FVZ - 1);
  const int xbase   = chunk * XCHUNK;
  const int rowbase = tid >> 6;           // 0..15

  float acc[16];
#pragma unroll
  for (int i = 0; i < 16; ++i) acc[i] = 0.0f;

  // Prime the TDM pipeline: plane for view 0 into buffer 0.
  if (tid == 0) tdm_load_plane(hmb, lds);

  for (int v = 0; v < NVIEW; ++v) {
    if (tid == 0) {
      if (v + 1 < NVIEW) {
        // Double-buffer: kick off view v+1 while we consume view v.
        tdm_load_plane(hmb + (u64)(v + 1) * NJ * PLANE, lds + ((v + 1) & 1) * PLANE);
        __builtin_amdgcn_s_wait_tensorcnt(1);   // TDM is in-order: view v's copy is done
      } else {
        __builtin_amdgcn_s_wait_tensorcnt(0);
      }
    }
    __syncthreads();                            // plane v visible to all waves

    const float* hmL = lds + (v & 1) * PLANE;
    const float* gv  = grids + (u64)v * (FVX * FVY * FVZ * 2);

#pragma unroll
    for (int it = 0; it < 16; ++it) {
      const int yl = rowbase + it * 16;         // 0..255
      const int y  = yl & 63;
      const int x  = (yl >> 6) + xbase;
      const int fxc = imin(imax(cx + x, 0), FVX - 1);
      const int fyc = imin(imax(cy + y, 0), FVY - 1);
      // contiguous along z across lanes -> coalesced b64 loads
      const int gi = ((fxc * FVY + fyc) * FVZ + fzc) * 2;
      const float2 g2 = *reinterpret_cast<const float2*>(gv + gi);

      const float ix = (g2.x + 1.0f) * 0.5f * (float)(WW - 1);
      const float iy = (g2.y + 1.0f) * 0.5f * (float)(HH - 1);
      const float x0f = floorf(ix), y0f = floorf(iy);
      const float wx = ix - x0f, wy = iy - y0f;
      const int x0 = (int)x0f, y0 = (int)y0f;
      const int x1 = x0 + 1,   y1 = y0 + 1;
      const int x0c = imin(imax(x0, 0), WW - 1), x1c = imin(imax(x1, 0), WW - 1);
      const int y0c = imin(imax(y0, 0), HH - 1), y1c = imin(imax(y1, 0), HH - 1);
      const float vx0 = (x0 >= 0 && x0 < WW) ? 1.0f : 0.0f;
      const float vx1 = (x1 < WW) ? 1.0f : 0.0f;      // x1 >= 1 > 0 always
      const float vy0 = (y0 >= 0 && y0 < HH) ? 1.0f : 0.0f;
      const float vy1 = (y1 < HH) ? 1.0f : 0.0f;
      // 4 taps served from LDS (ds_load_b32) instead of global gathers
      const float t00 = hmL[y0c * WW + x0c] * (vx0 * vy0);
      const float t10 = hmL[y0c * WW + x1c] * (vx1 * vy0);
      const float t01 = hmL[y1c * WW + x0c] * (vx0 * vy1);
      const float t11 = hmL[y1c * WW + x1c] * (vx1 * vy1);
      acc[it] += t00 * (1.0f - wx) * (1.0f - wy)
               + t10 * wx * (1.0f - wy)
               + t01 * (1.0f - wx) * wy
               + t11 * wx * wy;
    }
    __syncthreads();   // all waves done reading buffer (v&1) before it is re-filled
  }

  // ---- finalize: average over views, mask by validity box, clamp, stream out ----
  const u64 obase = (u64)(p * NJ + j) * (64 * 64 * 64);
#pragma unroll
  for (int it = 0; it < 16; ++it) {
    const int yl = rowbase + it * 16;
    const int y  = yl & 63;
    const int x  = (yl >> 6) + xbase;
    const int fx = cx + x, fy = cy + y;
    const bool vxy = (fx >= sx) && (fx < ex) && (fy >= sy) && (fy < ey);
    float val = 0.0f;
    if (person_ok && vxy && vz) {
      val = acc[it] * (1.0f / (float)NVIEW);
      val = fminf(fmaxf(val, 0.0f), 1.0f);
    }
    __builtin_nontemporal_store(val, &out[obase + ((u64)x << 12) + (u64)(y << 6) + (u64)z]);
  }
}

extern "C" void kernel_launch(void* const* d_in, const int* in_sizes, int n_in,
                              void* d_out, int out_size, void* d_ws, size_t ws_size,
                              hipStream_t stream) {
  (void)in_sizes; (void)n_in; (void)out_size; (void)d_ws; (void)ws_size;
  const float* heatmaps = (const float*)d_in[0];
  const float* grids    = (const float*)d_in[1];
  const float* pcs      = (const float*)d_in[2];
  const int*   indexp   = (const int*)d_in[3];
  float*       out      = (float*)d_out;

  const size_t shmem = (size_t)2 * PLANE * sizeof(float);  // 245,760 B <= 320 KB WGP LDS
  // Opt in to >64KB dynamic LDS (deterministic, idempotent; not a stream op).
  (void)hipFuncSetAttribute((const void*)ProjectLayerTS_kernel,
                            hipFuncAttributeMaxDynamicSharedMemorySize, (int)shmem);

  dim3 grid(NP * NJ * XSPLIT);   // 2400 workgroups
  dim3 block(1024);              // 32 wave32 waves -> fills a WGP
  ProjectLayerTS_kernel<<<grid, block, shmem, stream>>>(heatmaps, grids, pcs, indexp, out);
}